// RotAttLayer_89962384982591
// MI455X (gfx1250) — compile-verified
//
#include <hip/hip_runtime.h>
#include <math.h>
#include <stdint.h>

#define N_ENT   100000
#define N_REL   500
#define IN_DIM  128
#define OUT_DIM 128
#define N_TRIP  250000
#define M_ROWS  (2*N_TRIP)   // 500000 rows, divisible by 16
#define K_DIM   (3*IN_DIM)   // 384
#define BN_EPS  1e-5f
#define LRELU   0.01f

typedef __attribute__((ext_vector_type(2))) float v2f;
typedef __attribute__((ext_vector_type(8))) float v8f;

// ---- workspace layout (float offsets) ----
#define WS_BN0SUM 0        // 384
#define WS_BN0SS  384      // 384
#define WS_SCALE0 768      // 384
#define WS_SHIFT0 1152     // 384
#define WS_BN1SUM 1536     // 128
#define WS_BN1SS  1664     // 128
#define WS_SCALE1 1792     // 128
#define WS_SHIFT1 1920     // 128
#define WS_BPRIME 2048     // 128
#define WS_WT     2176     // 384*128 = 49152, packed in WMMA-fragment order:
                           //   wt_pack[(k>>1)*256 + n*2 + (k&1)] = a_w[n][k]*scale0[k]
#define WS_EBS    51328    // N_ENT
#define WS_RELCNT 151328   // N_REL
#define WS_TOTAL  151840

#define SH_STRIDE 388      // 384 padded (mult of 4 floats, rows hit distinct banks)

// -------------------- zero accumulators --------------------
__global__ void k_zero(float* __restrict__ out, float* __restrict__ ws) {
    long idx = (long)blockIdx.x * blockDim.x + threadIdx.x;
    const long n_out = (long)N_ENT * OUT_DIM + (long)N_REL * OUT_DIM;
    if (idx < n_out) {
        out[idx] = 0.f;
    } else {
        long w = idx - n_out;
        if (w < WS_TOTAL) ws[w] = 0.f;
    }
}

// -------------------- BN0 column statistics --------------------
// fwd/bwd symmetry: cols [0,128) & [128,256) share stats; cols [256,384) zero-mean
__global__ void k_bn0_stats(const int* __restrict__ trip,
                            const float* __restrict__ ent,
                            const float* __restrict__ rel,
                            float* __restrict__ ws) {
    int j = threadIdx.x; // 0..127
    float s = 0.f, ss = 0.f, ssr = 0.f;
    for (int t = blockIdx.x; t < N_TRIP; t += gridDim.x) {
        int t0 = trip[t*3+0], t1 = trip[t*3+1], t2 = trip[t*3+2];
        float e0 = ent[(long)t0*IN_DIM + j];
        float e1 = ent[(long)t1*IN_DIM + j];
        float r  = rel[(long)t2*IN_DIM + j];
        s   += e0 + e1;
        ss  += e0*e0 + e1*e1;
        ssr += 2.f * r * r;
    }
    atomicAdd(&ws[WS_BN0SUM +       j], s);
    atomicAdd(&ws[WS_BN0SUM + 128 + j], s);
    atomicAdd(&ws[WS_BN0SS  +       j], ss);
    atomicAdd(&ws[WS_BN0SS  + 128 + j], ss);
    atomicAdd(&ws[WS_BN0SS  + 256 + j], ssr);  // sum stays 0 for r/-r columns
}

__global__ void k_bn0_final(const float* __restrict__ g0,
                            const float* __restrict__ b0,
                            float* __restrict__ ws) {
    int j = threadIdx.x; // 384 threads
    float m  = ws[WS_BN0SUM + j] * (1.f / M_ROWS);
    float vv = ws[WS_BN0SS  + j] * (1.f / M_ROWS) - m * m;
    float sc = rsqrtf(vv + BN_EPS) * g0[j];
    ws[WS_SCALE0 + j] = sc;
    ws[WS_SHIFT0 + j] = b0[j] - m * sc;
}

// -------------------- fold BN0 into GEMM weight/bias --------------------
__global__ void k_fold(const float* __restrict__ aw,
                       const float* __restrict__ ab,
                       float* __restrict__ ws) {
    __shared__ float red[128];
    int n   = blockIdx.x;   // 0..127 output column
    int tid = threadIdx.x;  // 128 threads
    float p = 0.f;
    for (int k = tid; k < K_DIM; k += 128) {
        float w = aw[(long)n*K_DIM + k];
        // packed fragment order: lane-pair (k, k^1) adjacent -> b64 loads in GEMM
        ws[WS_WT + (long)(k >> 1) * 256 + n * 2 + (k & 1)] = w * ws[WS_SCALE0 + k];
        p += w * ws[WS_SHIFT0 + k];
    }
    red[tid] = p; __syncthreads();
    for (int o = 64; o > 0; o >>= 1) { if (tid < o) red[tid] += red[tid + o]; __syncthreads(); }
    if (tid == 0) ws[WS_BPRIME + n] = ab[n] + red[0];
}

// -------------------- stage 16 gathered rows into LDS --------------------
// entity halves: async global->LDS b128 copies (ASYNCcnt); relation third: manual (sign flip)
__device__ __forceinline__ void stage_rows(const int* __restrict__ trip,
                                           const float* __restrict__ ent,
                                           const float* __restrict__ rel,
                                           int rowBase, float* sh_h,
                                           int* sh_a, int* sh_b, int* sh_r,
                                           float* sh_sign, int* sh_fwd) {
    int tid = threadIdx.x;
    if (tid < 16) {
        int i = rowBase + tid;
        int a, b, r2, fw; float sg;
        if (i < N_TRIP) {             // fwd row: [e(t0), e(t1), +r]
            a = trip[i*3+0]; b = trip[i*3+1]; r2 = trip[i*3+2]; sg = 1.f; fw = 1;
        } else {                      // bwd row: [e(t1), e(t0), -r]
            int t = i - N_TRIP;
            int h = trip[t*3+0], tl = trip[t*3+1]; r2 = trip[t*3+2];
            a = tl; b = h; sg = -1.f; fw = 0;
        }
        sh_a[tid] = a; sh_b[tid] = b; sh_r[tid] = r2; sh_sign[tid] = sg; sh_fwd[tid] = fw;
    }
    __syncthreads();

    int lane = tid & 31;
    int wave = tid >> 5;
    // 32 row-copies (16 rows x 2 entity halves); each wave issues 4; each lane moves 16B
    #pragma unroll
    for (int q = 0; q < 4; ++q) {
        int c    = wave * 4 + q;
        int rr   = c >> 1;
        int part = c & 1;
        int erow = part ? sh_b[rr] : sh_a[rr];
        const void* src = (const void*)(ent + (long)erow * IN_DIM + lane * 4);
        unsigned lds_addr = (unsigned)(uintptr_t)(sh_h + rr * SH_STRIDE + part * 128 + lane * 4);
        asm volatile("global_load_async_to_lds_b128 %0, %1, off"
                     :: "v"(lds_addr), "v"(src) : "memory");
    }
    // relation third with sign applied (can't async-copy a negation)
    for (int idx = tid; idx < 16 * 128; idx += 256) {
        int rr = idx >> 7, j = idx & 127;
        sh_h[rr * SH_STRIDE + 256 + j] = sh_sign[rr] * rel[(long)sh_r[rr] * IN_DIM + j];
    }
    asm volatile("s_wait_asynccnt 0x0" ::: "memory");
    __syncthreads();
}

// -------------------- 16x16 tile via V_WMMA_F32_16X16X4_F32 --------------------
// A 16x4 frag: lane<16 -> row=lane, K={k0,k0+1}; lane>=16 -> row=lane-16, K={k0+2,k0+3}
// B 4x16 frag mirrored per column; wt is pre-packed so each lane does one b64 load
__device__ __forceinline__ v8f wmma_tile(const float* sh_h,
                                         const float* __restrict__ wtp,
                                         const float* __restrict__ bprime,
                                         int lane, int n0) {
    int row  = lane & 15;
    int g    = lane >> 4;              // 0 or 1
    int koff = g << 1;                 // 0 or 2
    int col  = n0 + (lane & 15);
    float bias = bprime[col];
    v8f c = { bias, bias, bias, bias, bias, bias, bias, bias };
    const float* hrow  = sh_h + row * SH_STRIDE;
    const float* bbase = wtp + (long)col * 2 + (long)g * 256; // pair p = k0/2 + g
    #pragma unroll 8
    for (int k0 = 0; k0 < K_DIM; k0 += 4) {
        v2f a, b;
        a.x = hrow[k0 + koff];
        a.y = hrow[k0 + koff + 1];
        b   = *(const v2f*)(bbase + (long)(k0 >> 1) * 256);
        c = __builtin_amdgcn_wmma_f32_16x16x4_f32(false, a, false, b,
                                                  (short)0, c, false, false);
    }
    return c;
}

// -------------------- GEMM pass 1: BN1 column stats --------------------
__global__ void __launch_bounds__(256)
k_gemm_stats(const int* __restrict__ trip, const float* __restrict__ ent,
             const float* __restrict__ rel, float* __restrict__ ws) {
    __shared__ float sh_h[16 * SH_STRIDE];
    __shared__ int   sh_a[16], sh_b[16], sh_r[16], sh_fwd[16];
    __shared__ float sh_sign[16];
    int rowBase = blockIdx.x * 16;
    stage_rows(trip, ent, rel, rowBase, sh_h, sh_a, sh_b, sh_r, sh_sign, sh_fwd);

    int lane = threadIdx.x & 31;
    int n0   = (threadIdx.x >> 5) * 16;   // 8 waves x 16 cols = 128
    v8f c = wmma_tile(sh_h, ws + WS_WT, ws + WS_BPRIME, lane, n0);

    float s = 0.f, ss = 0.f;
    #pragma unroll
    for (int v = 0; v < 8; ++v) { float x = c[v]; s += x; ss += x * x; }
    int col = n0 + (lane & 15);
    atomicAdd(&ws[WS_BN1SUM + col], s);
    atomicAdd(&ws[WS_BN1SS  + col], ss);
}

__global__ void k_bn1_final(const float* __restrict__ g1,
                            const float* __restrict__ b1,
                            float* __restrict__ ws) {
    int j = threadIdx.x; // 128
    float m  = ws[WS_BN1SUM + j] * (1.f / M_ROWS);
    float vv = ws[WS_BN1SS  + j] * (1.f / M_ROWS) - m * m;
    float sc = rsqrtf(vv + BN_EPS) * g1[j];
    ws[WS_SCALE1 + j] = sc;
    ws[WS_SHIFT1 + j] = b1[j] - m * sc;
}

// -------------------- GEMM pass 2: attention + segment scatter --------------------
__global__ void __launch_bounds__(256)
k_gemm_scatter(const int* __restrict__ trip, const float* __restrict__ ent,
               const float* __restrict__ rel, const float* __restrict__ a2w,
               const float* __restrict__ a2b, float* __restrict__ ws,
               float* __restrict__ out) {
    __shared__ float sh_h[16 * SH_STRIDE];
    __shared__ int   sh_a[16], sh_b[16], sh_r[16], sh_fwd[16];
    __shared__ float sh_sign[16];
    __shared__ float sh_dotw[8 * 16];     // per-wave row-dot partials
    __shared__ float sh_eb[16];
    int tid = threadIdx.x;
    if (tid < 128) sh_dotw[tid] = 0.f;    // visible after stage_rows' barriers
    int rowBase = blockIdx.x * 16;
    stage_rows(trip, ent, rel, rowBase, sh_h, sh_a, sh_b, sh_r, sh_sign, sh_fwd);

    int lane = tid & 31;
    int wave = tid >> 5;
    int n0   = wave * 16;
    int col  = n0 + (lane & 15);
    v8f cp = wmma_tile(sh_h, ws + WS_WT, ws + WS_BPRIME, lane, n0);

    float sc1 = ws[WS_SCALE1 + col], sf1 = ws[WS_SHIFT1 + col];
    float w2  = a2w[col];
    float cc[8];
    #pragma unroll
    for (int v = 0; v < 8; ++v) cc[v] = cp[v] * sc1 + sf1;   // BN1-normalized c

    // per-row dot partial: reduce over the 16 lanes of each half-wave via shfl_xor
    int mbase = (lane >> 4) * 8;          // C rows: v (+8 for hi lanes)
    float dp[8];
    #pragma unroll
    for (int v = 0; v < 8; ++v) dp[v] = cc[v] * w2;
    #pragma unroll
    for (int o = 8; o >= 1; o >>= 1) {
        #pragma unroll
        for (int v = 0; v < 8; ++v) dp[v] += __shfl_xor(dp[v], o, 32);
    }
    if ((lane & 15) == 0) {
        #pragma unroll
        for (int v = 0; v < 8; ++v) sh_dotw[wave * 16 + mbase + v] = dp[v];
    }
    __syncthreads();

    if (tid < 16) {
        float d = a2b[0];
        #pragma unroll
        for (int w = 0; w < 8; ++w) d += sh_dotw[w * 16 + tid];
        float l  = (d >= 0.f) ? d : LRELU * d;
        float eb = expf(-l);              // e_b = exp(-leaky_relu(.))
        sh_eb[tid] = eb;
        atomicAdd(&ws[WS_EBS + sh_a[tid]], eb);
        if (sh_fwd[tid]) atomicAdd(&ws[WS_RELCNT + sh_r[tid]], 1.f);
    }
    __syncthreads();

    float* hent = out;
    float* hrel = out + (long)N_ENT * OUT_DIM;
    #pragma unroll
    for (int v = 0; v < 8; ++v) {
        int m = mbase + v;
        float t1 = sh_eb[m] * cc[v];
        atomicAdd(&hent[(long)sh_a[m] * OUT_DIM + col], t1);
        if (sh_fwd[m]) atomicAdd(&hrel[(long)sh_r[m] * OUT_DIM + col], t1);
    }
}

// -------------------- finalize: divide accumulators --------------------
__global__ void k_final(float* __restrict__ out, const float* __restrict__ ws) {
    long idx = (long)blockIdx.x * blockDim.x + threadIdx.x;
    const long n_ent_el = (long)N_ENT * OUT_DIM;
    const long n_tot    = n_ent_el + (long)N_REL * OUT_DIM;
    if (idx >= n_tot) return;
    if (idx < n_ent_el) {
        int row = (int)(idx / OUT_DIM);
        float d = ws[WS_EBS + row];
        d = (d == 0.f) ? 1e-12f : d;
        out[idx] = out[idx] / d;
    } else {
        long r = idx - n_ent_el;
        int row = (int)(r / OUT_DIM);
        out[idx] = out[idx] / fmaxf(ws[WS_RELCNT + row], 1.f);
    }
}

extern "C" void kernel_launch(void* const* d_in, const int* in_sizes, int n_in,
                              void* d_out, int out_size, void* d_ws, size_t ws_size,
                              hipStream_t stream) {
    const int*   trip = (const int*)  d_in[0];
    const float* ent  = (const float*)d_in[1];
    const float* rel  = (const float*)d_in[2];
    const float* aw   = (const float*)d_in[3];
    const float* ab   = (const float*)d_in[4];
    const float* a2w  = (const float*)d_in[5];
    const float* a2b  = (const float*)d_in[6];
    const float* g0   = (const float*)d_in[7];
    const float* b0   = (const float*)d_in[8];
    const float* g1   = (const float*)d_in[9];
    const float* b1   = (const float*)d_in[10];
    float* out = (float*)d_out;
    float* ws  = (float*)d_ws;

    {
        long total = (long)N_ENT * OUT_DIM + (long)N_REL * OUT_DIM + WS_TOTAL;
        int blocks = (int)((total + 255) / 256);
        k_zero<<<blocks, 256, 0, stream>>>(out, ws);
    }
    k_bn0_stats<<<1024, 128, 0, stream>>>(trip, ent, rel, ws);
    k_bn0_final<<<1, 384, 0, stream>>>(g0, b0, ws);
    k_fold<<<128, 128, 0, stream>>>(aw, ab, ws);
    k_gemm_stats<<<M_ROWS / 16, 256, 0, stream>>>(trip, ent, rel, ws);
    k_bn1_final<<<1, 128, 0, stream>>>(g1, b1, ws);
    k_gemm_scatter<<<M_ROWS / 16, 256, 0, stream>>>(trip, ent, rel, a2w, a2b, ws, out);
    {
        long total = (long)N_ENT * OUT_DIM + (long)N_REL * OUT_DIM;
        int blocks = (int)((total + 255) / 256);
        k_final<<<blocks, 256, 0, stream>>>(out, ws);
    }
}